// MambaBlock_18262200943213
// MI455X (gfx1250) — compile-verified
//
#include <hip/hip_runtime.h>
#include <hip/hip_bf16.h>
#include <math.h>

// ---------------- model dims (match reference) ----------------
#define D_MODEL 1024
#define D_STATE 16
#define D_CONV  4
#define D_INNER 2048            // EXPAND * D_MODEL
#define DT_RANK 64              // D_MODEL / 16
#define B_SZ    2
#define L_SEQ   2048
#define NTOK    (B_SZ * L_SEQ)  // 4096 tokens
#define XP_COLS (DT_RANK + 2 * D_STATE)  // 96

typedef float v2f __attribute__((ext_vector_type(2)));
typedef float v8f __attribute__((ext_vector_type(8)));

// D = A(16x4,f32) x B(4x16,f32) + C(16x16,f32), wave32 WMMA
__device__ __forceinline__ v8f wmma_f32_k4(v2f a, v2f b, v8f c) {
    return __builtin_amdgcn_wmma_f32_16x16x4_f32(
        /*neg_a=*/false, a, /*neg_b=*/false, b,
        /*c_mod=*/(short)0, c, /*reuse_a=*/false, /*reuse_b=*/false);
}

__device__ __forceinline__ float silu_f(float v) {
    return v / (1.0f + __expf(-v));
}

// ---------------- 1) pre-LayerNorm over D_MODEL per token ----------------
__global__ __launch_bounds__(256)
void ln_kernel(const float* __restrict__ x, const float* __restrict__ g,
               const float* __restrict__ bta, float* __restrict__ xn)
{
    const int tok = blockIdx.x;
    const int tid = threadIdx.x;
    const float* row = x + (size_t)tok * D_MODEL;

    float v[4];
    float s = 0.f;
#pragma unroll
    for (int i = 0; i < 4; ++i) { v[i] = row[tid + 256 * i]; s += v[i]; }

    __shared__ float red[256];
    red[tid] = s;
    __syncthreads();
    for (int off = 128; off > 0; off >>= 1) {
        if (tid < off) red[tid] += red[tid + off];
        __syncthreads();
    }
    const float mu = red[0] * (1.0f / D_MODEL);
    __syncthreads();

    float vs = 0.f;
#pragma unroll
    for (int i = 0; i < 4; ++i) { float d = v[i] - mu; vs += d * d; }
    red[tid] = vs;
    __syncthreads();
    for (int off = 128; off > 0; off >>= 1) {
        if (tid < off) red[tid] += red[tid + off];
        __syncthreads();
    }
    const float rstd = rsqrtf(red[0] * (1.0f / D_MODEL) + 1e-5f);

    float* orow = xn + (size_t)tok * D_MODEL;
#pragma unroll
    for (int i = 0; i < 4; ++i) {
        const int c = tid + 256 * i;
        orow[c] = (v[i] - mu) * rstd * g[c] + bta[c];
    }
}

// ---------------- f32 WMMA GEMM: C[m][n] = sum_k A[m][k]*B[n][k] ----------
// Wide variant: N multiple of 128, M multiple of 128, K multiple of 8.
// Block = 8 waves arranged 4(M) x 2(N); wave tile = 32(M) x 64(N):
//   2 A-fragments x 4 B-fragments -> 8 WMMAs per 6 loads, software-pipelined.
__global__ __launch_bounds__(256)
void gemm_wmma_n64(const float* __restrict__ A, int lda,
                   const float* __restrict__ B, int ldb,
                   float* __restrict__ C, int ldc,
                   const float* __restrict__ resid, int K)
{
    const int lane = threadIdx.x & 31;
    const int wave = threadIdx.x >> 5;
    const int m0   = blockIdx.y * 128 + (wave & 3) * 32;
    const int n0   = blockIdx.x * 128 + (wave >> 2) * 64;
    const int half = lane >> 4;     // 0: K+0/1, 1: K+2/3
    const int lr   = lane & 15;

    v8f acc00 = {}, acc01 = {}, acc02 = {}, acc03 = {};
    v8f acc10 = {}, acc11 = {}, acc12 = {}, acc13 = {};

    const float* Ap0 = A + (size_t)(m0 + lr) * lda + 2 * half;
    const float* Ap1 = Ap0 + (size_t)16 * lda;
    const float* B0  = B + (size_t)(n0 + lr) * ldb + 2 * half;
    const float* B1  = B0 + (size_t)16 * ldb;
    const float* B2  = B0 + (size_t)32 * ldb;
    const float* B3  = B0 + (size_t)48 * ldb;

    // speculative prefetch of the first K-chunk ahead (global_prefetch_b8)
    __builtin_prefetch(Ap0 + 64, 0, 1);
    __builtin_prefetch(Ap1 + 64, 0, 1);
    __builtin_prefetch(B0 + 64, 0, 1);
    __builtin_prefetch(B1 + 64, 0, 1);
    __builtin_prefetch(B2 + 64, 0, 1);
    __builtin_prefetch(B3 + 64, 0, 1);

    // software pipeline: fragments for step k live while step k+4 loads issue
    v2f a0 = *(const v2f*)Ap0;
    v2f a1 = *(const v2f*)Ap1;
    v2f b0 = *(const v2f*)B0;
    v2f b1 = *(const v2f*)B1;
    v2f b2 = *(const v2f*)B2;
    v2f b3 = *(const v2f*)B3;

#pragma unroll 8
    for (int k = 4; k < K; k += 4) {
        v2f na0 = *(const v2f*)(Ap0 + k);
        v2f na1 = *(const v2f*)(Ap1 + k);
        v2f nb0 = *(const v2f*)(B0 + k);
        v2f nb1 = *(const v2f*)(B1 + k);
        v2f nb2 = *(const v2f*)(B2 + k);
        v2f nb3 = *(const v2f*)(B3 + k);

        acc00 = wmma_f32_k4(a0, b0, acc00);
        acc01 = wmma_f32_k4(a0, b1, acc01);
        acc02 = wmma_f32_k4(a0, b2, acc02);
        acc03 = wmma_f32_k4(a0, b3, acc03);
        acc10 = wmma_f32_k4(a1, b0, acc10);
        acc11 = wmma_f32_k4(a1, b1, acc11);
        acc12 = wmma_f32_k4(a1, b2, acc12);
        acc13 = wmma_f32_k4(a1, b3, acc13);

        a0 = na0; a1 = na1;
        b0 = nb0; b1 = nb1; b2 = nb2; b3 = nb3;
    }
    // drain last stage
    acc00 = wmma_f32_k4(a0, b0, acc00);
    acc01 = wmma_f32_k4(a0, b1, acc01);
    acc02 = wmma_f32_k4(a0, b2, acc02);
    acc03 = wmma_f32_k4(a0, b3, acc03);
    acc10 = wmma_f32_k4(a1, b0, acc10);
    acc11 = wmma_f32_k4(a1, b1, acc11);
    acc12 = wmma_f32_k4(a1, b2, acc12);
    acc13 = wmma_f32_k4(a1, b3, acc13);

    // C/D layout: VGPR r -> row m0 + r + 8*half (+16 for second A tile),
    // col n0 + lr (+16 per B tile)
#pragma unroll
    for (int r = 0; r < 8; ++r) {
        const size_t base0 = (size_t)(m0 + r + 8 * half) * ldc + n0 + lr;
        const size_t base1 = base0 + (size_t)16 * ldc;
        float o00 = acc00[r], o01 = acc01[r], o02 = acc02[r], o03 = acc03[r];
        float o10 = acc10[r], o11 = acc11[r], o12 = acc12[r], o13 = acc13[r];
        if (resid) {
            o00 += resid[base0];      o01 += resid[base0 + 16];
            o02 += resid[base0 + 32]; o03 += resid[base0 + 48];
            o10 += resid[base1];      o11 += resid[base1 + 16];
            o12 += resid[base1 + 32]; o13 += resid[base1 + 48];
        }
        C[base0]      = o00; C[base0 + 16] = o01;
        C[base0 + 32] = o02; C[base0 + 48] = o03;
        C[base1]      = o10; C[base1 + 16] = o11;
        C[base1 + 32] = o12; C[base1 + 48] = o13;
    }
}

// Narrow variant: N multiple of 16 (for the 96-wide x_dbl projection).
// Block = 8 waves along M -> block tile 128(M) x 16(N); wave tile 16x16.
__global__ __launch_bounds__(256)
void gemm_wmma_n16(const float* __restrict__ A, int lda,
                   const float* __restrict__ B, int ldb,
                   float* __restrict__ C, int ldc, int K)
{
    const int lane = threadIdx.x & 31;
    const int wave = threadIdx.x >> 5;
    const int m0   = blockIdx.y * 128 + wave * 16;
    const int n0   = blockIdx.x * 16;
    const int half = lane >> 4;
    const int lr   = lane & 15;

    v8f acc = {};
    const float* Ap = A + (size_t)(m0 + lr) * lda + 2 * half;
    const float* Bp = B + (size_t)(n0 + lr) * ldb + 2 * half;

    __builtin_prefetch(Ap + 64, 0, 1);
    __builtin_prefetch(Bp + 64, 0, 1);

    v2f a = *(const v2f*)Ap;
    v2f b = *(const v2f*)Bp;

#pragma unroll 8
    for (int k = 4; k < K; k += 4) {
        v2f na = *(const v2f*)(Ap + k);
        v2f nb = *(const v2f*)(Bp + k);
        acc = wmma_f32_k4(a, b, acc);
        a = na; b = nb;
    }
    acc = wmma_f32_k4(a, b, acc);

#pragma unroll
    for (int r = 0; r < 8; ++r) {
        C[(size_t)(m0 + r + 8 * half) * ldc + n0 + lr] = acc[r];
    }
}

// ---------------- 3) causal depthwise conv (d_conv=4) + SiLU ----------------
// u = xz[:, :, 0:D_INNER];  u_c[b,t,d] = silu(bias + sum_j u[b,t-3+j,d]*w[d,j])
__global__ __launch_bounds__(256)
void conv_silu_kernel(const float* __restrict__ xz,
                      const float* __restrict__ cw,
                      const float* __restrict__ cb,
                      float* __restrict__ uc)
{
    const size_t idx = (size_t)blockIdx.x * blockDim.x + threadIdx.x;
    if (idx >= (size_t)NTOK * D_INNER) return;
    const int d  = (int)(idx & (D_INNER - 1));
    const size_t bt = idx >> 11;              // log2(D_INNER) = 11
    const int t  = (int)(bt & (L_SEQ - 1));
    const int b  = (int)(bt >> 11);           // log2(L_SEQ) = 11

    float acc = cb[d];
#pragma unroll
    for (int j = 0; j < D_CONV; ++j) {
        const int tt = t - (D_CONV - 1) + j;
        if (tt >= 0) {
            acc += xz[((size_t)(b * L_SEQ + tt)) * (2 * D_INNER) + d] * cw[d * D_CONV + j];
        }
    }
    uc[idx] = silu_f(acc);
}

// ---------------- 5b) dt = softplus(dt_raw + b_dt), in place ----------------
__global__ __launch_bounds__(256)
void softplus_kernel(const float* __restrict__ raw,
                     const float* __restrict__ b_dt,
                     float* __restrict__ dt)
{
    const size_t idx = (size_t)blockIdx.x * blockDim.x + threadIdx.x;
    if (idx >= (size_t)NTOK * D_INNER) return;
    const int d = (int)(idx & (D_INNER - 1));
    const float v = raw[idx] + b_dt[d];
    dt[idx] = (v > 20.f) ? v : log1pf(__expf(v));
}

// ---------------- 6) selective scan + D skip + SiLU(z) gating ----------------
// One thread per (b, d); 16-state recurrence in registers; B_t/C_t broadcast via LDS.
__global__ __launch_bounds__(256)
void scan_kernel(const float* __restrict__ uc,
                 const float* dtbuf,            // not restrict: may alias y
                 const float* __restrict__ xdbl,
                 const float* __restrict__ A_log,
                 const float* __restrict__ Dvec,
                 const float* __restrict__ xz,
                 float* ybuf)                   // may alias dtbuf
{
    const int b = blockIdx.x >> 3;                      // D_INNER/256 = 8 chunks
    const int d = ((blockIdx.x & 7) << 8) + threadIdx.x;

    float Ad[D_STATE];
#pragma unroll
    for (int n = 0; n < D_STATE; ++n)
        Ad[n] = -__expf(A_log[(size_t)d * D_STATE + n]);
    const float Dd = Dvec[d];

    float h[D_STATE];
#pragma unroll
    for (int n = 0; n < D_STATE; ++n) h[n] = 0.f;

    __shared__ float sBC[2 * D_STATE];                  // [0:16)=B_t, [16:32)=C_t

    for (int t = 0; t < L_SEQ; ++t) {
        const size_t tok = (size_t)b * L_SEQ + t;
        if (threadIdx.x < 2 * D_STATE)
            sBC[threadIdx.x] = xdbl[tok * XP_COLS + DT_RANK + threadIdx.x];
        __syncthreads();

        const size_t i = tok * D_INNER + d;
        const float u   = uc[i];
        const float dtv = dtbuf[i];
        const float du  = dtv * u;
        float yv = 0.f;
#pragma unroll
        for (int n = 0; n < D_STATE; ++n) {
            h[n] = __expf(dtv * Ad[n]) * h[n] + du * sBC[n];
            yv  += h[n] * sBC[D_STATE + n];
        }
        yv += u * Dd;

        const float z = xz[tok * (2 * D_INNER) + D_INNER + d];
        ybuf[i] = yv * silu_f(z);                       // reads dt[i] happened above
        __syncthreads();
    }
}

// ---------------- launcher ----------------
extern "C" void kernel_launch(void* const* d_in, const int* in_sizes, int n_in,
                              void* d_out, int out_size, void* d_ws, size_t ws_size,
                              hipStream_t stream)
{
    const float* x      = (const float*)d_in[0];
    const float* ln_g   = (const float*)d_in[1];
    const float* ln_b   = (const float*)d_in[2];
    const float* W_in   = (const float*)d_in[3];   // (2*D_INNER, D_MODEL)
    const float* conv_w = (const float*)d_in[4];   // (D_INNER, D_CONV)
    const float* conv_b = (const float*)d_in[5];
    const float* W_xp   = (const float*)d_in[6];   // (96, D_INNER)
    const float* W_dt   = (const float*)d_in[7];   // (D_INNER, DT_RANK)
    const float* b_dt   = (const float*)d_in[8];
    const float* A_log  = (const float*)d_in[9];   // (D_INNER, D_STATE)
    const float* Dv     = (const float*)d_in[10];
    const float* W_out  = (const float*)d_in[11];  // (D_MODEL, D_INNER)
    float* out = (float*)d_out;

    // workspace partition (floats): ~153 MB total
    float* xn   = (float*)d_ws;                               // 4096*1024
    float* xz   = xn   + (size_t)NTOK * D_MODEL;              // 4096*4096
    float* uc   = xz   + (size_t)NTOK * 2 * D_INNER;          // 4096*2048
    float* xdbl = uc   + (size_t)NTOK * D_INNER;              // 4096*96
    float* dt   = xdbl + (size_t)NTOK * XP_COLS;              // 4096*2048
    float* y    = dt;  // alias: scan reads dt[i] before writing y[i] (same thread)

    // 1) LayerNorm
    ln_kernel<<<NTOK, 256, 0, stream>>>(x, ln_g, ln_b, xn);

    // 2) xz = xn @ W_in^T   (M=4096, N=4096, K=1024)
    gemm_wmma_n64<<<dim3((2 * D_INNER) / 128, NTOK / 128), 256, 0, stream>>>(
        xn, D_MODEL, W_in, D_MODEL, xz, 2 * D_INNER, nullptr, D_MODEL);

    // 3) u_c = silu(depthwise_causal_conv(u) + b)
    conv_silu_kernel<<<(NTOK * D_INNER) / 256, 256, 0, stream>>>(xz, conv_w, conv_b, uc);

    // 4) x_dbl = u_c @ W_xp^T   (M=4096, N=96, K=2048)
    gemm_wmma_n16<<<dim3(XP_COLS / 16, NTOK / 128), 256, 0, stream>>>(
        uc, D_INNER, W_xp, D_INNER, xdbl, XP_COLS, D_INNER);

    // 5) dt_raw = x_dbl[:, :64] @ W_dt^T   (M=4096, N=2048, K=64)
    gemm_wmma_n64<<<dim3(D_INNER / 128, NTOK / 128), 256, 0, stream>>>(
        xdbl, XP_COLS, W_dt, DT_RANK, dt, D_INNER, nullptr, DT_RANK);

    // 5b) dt = softplus(dt_raw + b_dt)
    softplus_kernel<<<(NTOK * D_INNER) / 256, 256, 0, stream>>>(dt, b_dt, dt);

    // 6) selective scan + D skip + silu(z) gating -> y
    scan_kernel<<<B_SZ * (D_INNER / 256), 256, 0, stream>>>(
        uc, dt, xdbl, A_log, Dv, xz, y);

    // 7) out = x + y @ W_out^T   (M=4096, N=1024, K=2048), residual fused
    gemm_wmma_n64<<<dim3(D_MODEL / 128, NTOK / 128), 256, 0, stream>>>(
        y, D_INNER, W_out, D_INNER, out, D_MODEL, x, D_INNER);
}